// ComplexAgnosticResidualInteractionBlock_8675833938671
// MI455X (gfx1250) — compile-verified
//
#include <hip/hip_runtime.h>
#include <hip/hip_bf16.h>

// ---------------------------------------------------------------------------
// ComplexAgnosticResidualInteractionBlock for MI455X (gfx1250, wave32, WMMA)
//
// N=10000 E=128000 C=128 A=10 F=8  LDIMS=(1,3,5)  MSG=C*9=1152
//
// d_ws layout (floats):
//   h      [N,128]        = node_feats @ W_up
//   msg_r  [N,1152]       component-major: [n][comp(9)][u(128)]
//   msg_i  [N,1152]
//   packed weights        fragment-packed B operands (built per launch)
// d_out layout (floats): out_r [N,1152] | out_i [N,1152] | sc [N,128]
// ---------------------------------------------------------------------------

#define N_NODES 10000
#define E_EDGES 128000
#define C_DIM   128
#define A_DIM   10
#define F_DIM   8
#define MSG_DIM 1152

typedef __attribute__((ext_vector_type(2))) float v2f;
typedef __attribute__((ext_vector_type(8))) float v8f;

__device__ __forceinline__ v8f wmma4(v2f a, v2f b, v8f c) {
  // V_WMMA_F32_16X16X4_F32 : D = A(16x4) * B(4x16) + C(16x16), fp32
  return __builtin_amdgcn_wmma_f32_16x16x4_f32(
      false, a, false, b, (short)0, c, false, false);
}

__device__ __forceinline__ float silu(float x) {
  return x / (1.0f + __expf(-x));
}

// B-fragment from fragment-packed weights: one coalesced b64 per WMMA.
// packed[(k>>1)*2N + 2n + (k&1)]; lane pair index = (k0>>1)+hl.
__device__ __forceinline__ v2f load_bfrag(const float* __restrict__ Wp, int N2,
                                          int k0, int hl, int ncol) {
  const float2 t = *(const float2*)(Wp + ((k0 >> 1) + hl) * N2 + ncol * 2);
  v2f b; b.x = t.x; b.y = t.y;
  return b;
}

// ---------------------------------------------------------------------------
// Kernel 0a: zero the atomic accumulation scratch
// ---------------------------------------------------------------------------
__global__ void zero_kernel(float* __restrict__ p, size_t n) {
  size_t i = (size_t)blockIdx.x * blockDim.x + threadIdx.x;
  size_t stride = (size_t)gridDim.x * blockDim.x;
  for (; i < n; i += stride) p[i] = 0.0f;
}

// ---------------------------------------------------------------------------
// Kernel 0b: fragment-pack a KxN weight (row stride srcStride) for B operands
// ---------------------------------------------------------------------------
__global__ void pack_b_kernel(const float* __restrict__ src, float* __restrict__ dst,
                              int K, int N, int srcStride) {
  int total = K * N;
  for (int i = blockIdx.x * blockDim.x + threadIdx.x; i < total;
       i += gridDim.x * blockDim.x) {
    int k = i / N, n = i - k * N;
    dst[(k >> 1) * (2 * N) + n * 2 + (k & 1)] = src[k * srcStride + n];
  }
}

// ---------------------------------------------------------------------------
// Kernel 1: node-side GEMMs (h and sc). Block = 256 = 8 waves; block owns a
// 16-row tile; wave w owns 16-wide column tile w. 11 GEMMs x 32 k-steps.
// ---------------------------------------------------------------------------
__global__ __launch_bounds__(256) void node_gemm_kernel(
    const float* __restrict__ node_attrs,   // [N,10]
    const float* __restrict__ node_feats,   // [N,128]
    const float* __restrict__ Wup_p,        // packed [128,128]
    const float* __restrict__ Wskip_p,      // packed, 10 x [128,128]
    float* __restrict__ h_out,              // [N,128]
    float* __restrict__ sc_out)             // [N,128]
{
  __shared__ float f_lds[16][C_DIM];
  __shared__ float a_lds[16][A_DIM];

  const int row0 = blockIdx.x * 16;
  const int tid  = threadIdx.x;

  for (int i = tid; i < 16 * C_DIM; i += 256) {
    int rr = i >> 7, cc = i & 127;
    f_lds[rr][cc] = node_feats[(row0 + rr) * C_DIM + cc];
  }
  for (int i = tid; i < 16 * A_DIM; i += 256) {
    int rr = i / A_DIM, cc = i % A_DIM;
    a_lds[rr][cc] = node_attrs[(row0 + rr) * A_DIM + cc];
  }
  __syncthreads();

  const int wave = tid >> 5;
  const int lane = tid & 31;
  const int hl   = lane >> 4;
  const int r    = lane & 15;
  const int col0 = wave * 16;
  const int nc   = col0 + r;

  v8f acc_h = {};
  for (int k0 = 0; k0 < C_DIM; k0 += 4) {
    const int ku = k0 + 2 * hl;
    v2f a;
    a.x = f_lds[r][ku];
    a.y = f_lds[r][ku + 1];
    acc_h = wmma4(a, load_bfrag(Wup_p, 256, k0, hl, nc), acc_h);
  }

  v8f acc_sc = {};
  for (int v = 0; v < A_DIM; ++v) {
    const float* Wv = Wskip_p + v * (C_DIM * C_DIM);
    v8f g = {};
    for (int k0 = 0; k0 < C_DIM; k0 += 4) {
      const int ku = k0 + 2 * hl;
      v2f a;
      a.x = f_lds[r][ku];
      a.y = f_lds[r][ku + 1];
      g = wmma4(a, load_bfrag(Wv, 256, k0, hl, nc), g);
    }
#pragma unroll
    for (int j = 0; j < 8; ++j) {
      const int m = j + 8 * hl;
      acc_sc[j] += a_lds[m][v] * g[j];
    }
  }

#pragma unroll
  for (int j = 0; j < 8; ++j) {
    const int n = row0 + j + 8 * hl;
    h_out[n * C_DIM + nc]  = acc_h[j];
    sc_out[n * C_DIM + nc] = acc_sc[j];
  }
}

// ---------------------------------------------------------------------------
// Kernel 2: fused edge pipeline. Block = 128 = 4 waves; each wave owns 16
// edges. 4-layer MLP in WMMA with ping/pong per-wave LDS activations; the
// [16,384] tpw tile stays in registers and feeds the uvu tensor product +
// global_atomic_add_f32 segment-sum directly (mji/tpw never hit HBM).
// msg layout is component-major: msg[n, comp*128 + c].
// ---------------------------------------------------------------------------
__global__ __launch_bounds__(128) void edge_kernel(
    const float* __restrict__ edge_attrs_real,  // [E,9]
    const float* __restrict__ edge_attrs_imag,  // [E,9]
    const float* __restrict__ edge_feats,       // [E,8]
    const int*   __restrict__ edge_index,       // [2,E]
    const float* __restrict__ W1p,              // packed [8,64]
    const float* __restrict__ W2p,              // packed [64,64]
    const float* __restrict__ W3p,              // packed [64,64]
    const float* __restrict__ W4p,              // packed [64,384]
    const float* __restrict__ h,                // [N,128]
    float* __restrict__ msg_r,                  // [N,1152]
    float* __restrict__ msg_i)                  // [N,1152]
{
  __shared__ float feats[64][F_DIM];
  __shared__ float ysr[64][9];
  __shared__ float ysi[64][9];
  __shared__ int   snd[64];
  __shared__ int   rcv[64];
  __shared__ float tAb[4][16][64];
  __shared__ float tBb[4][16][64];

  const int eBase = blockIdx.x * 64;
  const int tid   = threadIdx.x;

  // warm the hot W4 block in cache (gfx1250 global_prefetch path)
  __builtin_prefetch(W4p + tid * 192, 0, 3);

  for (int i = tid; i < 64 * F_DIM; i += 128) {
    int e = i >> 3, c = i & 7;
    feats[e][c] = edge_feats[(eBase + e) * F_DIM + c];
  }
  for (int i = tid; i < 64 * 9; i += 128) {
    int e = i / 9, c = i % 9;
    ysr[e][c] = edge_attrs_real[(eBase + e) * 9 + c];
    ysi[e][c] = edge_attrs_imag[(eBase + e) * 9 + c];
  }
  for (int i = tid; i < 64; i += 128) {
    snd[i] = edge_index[eBase + i];
    rcv[i] = edge_index[E_EDGES + eBase + i];
  }
  __syncthreads();

  const int wave = tid >> 5;
  const int lane = tid & 31;
  const int hl   = lane >> 4;
  const int r    = lane & 15;
  const int ew0  = wave * 16;
  float (*tA)[64] = tAb[wave];
  float (*tB)[64] = tBb[wave];

  // ---- layer 1: [16,8] @ [8,64], silu -> tA -------------------------------
  for (int nt = 0; nt < 4; ++nt) {
    v8f acc = {};
    for (int k0 = 0; k0 < F_DIM; k0 += 4) {
      const int ku = k0 + 2 * hl;
      v2f a;
      a.x = feats[ew0 + r][ku];
      a.y = feats[ew0 + r][ku + 1];
      acc = wmma4(a, load_bfrag(W1p, 128, k0, hl, nt * 16 + r), acc);
    }
#pragma unroll
    for (int j = 0; j < 8; ++j)
      tA[j + 8 * hl][nt * 16 + r] = silu(acc[j]);
  }
  // ---- layer 2: tA @ W2, silu -> tB ---------------------------------------
  for (int nt = 0; nt < 4; ++nt) {
    v8f acc = {};
    for (int k0 = 0; k0 < 64; k0 += 4) {
      const int ku = k0 + 2 * hl;
      v2f a;
      a.x = tA[r][ku];
      a.y = tA[r][ku + 1];
      acc = wmma4(a, load_bfrag(W2p, 128, k0, hl, nt * 16 + r), acc);
    }
#pragma unroll
    for (int j = 0; j < 8; ++j)
      tB[j + 8 * hl][nt * 16 + r] = silu(acc[j]);
  }
  // ---- layer 3: tB @ W3, silu -> tA ---------------------------------------
  for (int nt = 0; nt < 4; ++nt) {
    v8f acc = {};
    for (int k0 = 0; k0 < 64; k0 += 4) {
      const int ku = k0 + 2 * hl;
      v2f a;
      a.x = tB[r][ku];
      a.y = tB[r][ku + 1];
      acc = wmma4(a, load_bfrag(W3p, 128, k0, hl, nt * 16 + r), acc);
    }
#pragma unroll
    for (int j = 0; j < 8; ++j)
      tA[j + 8 * hl][nt * 16 + r] = silu(acc[j]);
  }

  // ---- layer 4 + uvu tensor product + atomic scatter ----------------------
  // tpw column n = nt*16 + r ; path = n>>7 ; channel c = n&127.
  // 16-wide tiles never straddle a path boundary.
  const int yOff[3] = {0, 1, 4};   // first sh component per path
  const int pD[3]   = {1, 3, 5};   // d per path

  for (int nt = 0; nt < 24; ++nt) {
    v8f acc = {};
    for (int k0 = 0; k0 < 64; k0 += 4) {
      const int ku = k0 + 2 * hl;
      v2f a;
      a.x = tA[r][ku];
      a.y = tA[r][ku + 1];
      acc = wmma4(a, load_bfrag(W4p, 768, k0, hl, nt * 16 + r), acc);
    }
    const int path = nt >> 3;
    const int d    = pD[path];
    const int c    = ((nt & 7) << 4) + r;
#pragma unroll
    for (int j = 0; j < 8; ++j) {
      const int el  = ew0 + j + 8 * hl;
      const float x = h[snd[el] * C_DIM + c];
      const float xw = x * acc[j];
      const int base = rcv[el] * MSG_DIM + (yOff[path]) * C_DIM + c;
      for (int comp = 0; comp < d; ++comp) {
        atomicAdd(&msg_r[base + comp * C_DIM], xw * ysr[el][yOff[path] + comp]);
        atomicAdd(&msg_i[base + comp * C_DIM], xw * ysi[el][yOff[path] + comp]);
      }
    }
  }
}

// ---------------------------------------------------------------------------
// Kernel 3: irreps linear on component-major msg.
//   out[n, off_l + k*d + m] = sum_u msg[n, lm*128 + u] * W_l[u,k]
// Block = 256 = 8 waves per 16-node row tile; 144 GEMM jobs round-robined.
// A and B fragments are both single b64 loads now.
// ---------------------------------------------------------------------------
__global__ __launch_bounds__(256) void linear_irreps_kernel(
    const float* __restrict__ msg_r,
    const float* __restrict__ msg_i,
    const float* __restrict__ L0p,   // packed [128,128]
    const float* __restrict__ L1p,
    const float* __restrict__ L2p,
    float* __restrict__ out)         // out_r | out_i
{
  const int row0 = blockIdx.x * 16;
  const int wave = threadIdx.x >> 5;
  const int lane = threadIdx.x & 31;
  const int hl   = lane >> 4;
  const int r    = lane & 15;

  const int pOff[3] = {0, 128, 512};
  const int pD[3]   = {1, 3, 5};
  const int lmL[9]  = {0, 1, 1, 1, 2, 2, 2, 2, 2};
  const int lmM[9]  = {0, 0, 1, 2, 0, 1, 2, 3, 4};
  const float* Wp[3] = {L0p, L1p, L2p};

  for (int job = wave; job < 144; job += 8) {
    const int ri  = job / 72;
    const int rem = job % 72;
    const int lm  = rem >> 3;          // component index 0..8
    const int kt  = rem & 7;
    const int l   = lmL[lm];
    const int m   = lmM[lm];
    const int d   = pD[l];
    const int off = pOff[l];
    const float* M = ri ? msg_i : msg_r;
    const float* W = Wp[l];

    const float* arow = M + (row0 + r) * MSG_DIM + lm * C_DIM;
    v8f acc = {};
    for (int k0 = 0; k0 < C_DIM; k0 += 4) {
      const int u = k0 + 2 * hl;
      const float2 av = *(const float2*)(arow + u);
      v2f a; a.x = av.x; a.y = av.y;
      acc = wmma4(a, load_bfrag(W, 256, k0, hl, kt * 16 + r), acc);
    }

    float* O = out + (size_t)ri * ((size_t)N_NODES * MSG_DIM);
#pragma unroll
    for (int j = 0; j < 8; ++j) {
      const int n = row0 + j + 8 * hl;
      const int k = kt * 16 + r;
      O[(size_t)n * MSG_DIM + off + k * d + m] = acc[j];
    }
  }
}

// ---------------------------------------------------------------------------
extern "C" void kernel_launch(void* const* d_in, const int* in_sizes, int n_in,
                              void* d_out, int out_size, void* d_ws, size_t ws_size,
                              hipStream_t stream) {
  (void)in_sizes; (void)n_in; (void)out_size; (void)ws_size;

  const float* node_attrs = (const float*)d_in[0];
  const float* node_feats = (const float*)d_in[1];
  const float* ear        = (const float*)d_in[2];
  const float* eai        = (const float*)d_in[3];
  const float* ef         = (const float*)d_in[4];
  const int*   eidx       = (const int*)  d_in[5];
  const float* W_up       = (const float*)d_in[6];
  const float* W_skip     = (const float*)d_in[7];
  const float* W_mlp1     = (const float*)d_in[8];
  const float* W_mlp2     = (const float*)d_in[9];
  const float* W_mlp3     = (const float*)d_in[10];
  const float* W_mlp4     = (const float*)d_in[11];
  const float* W_lin0     = (const float*)d_in[12];
  const float* W_lin1     = (const float*)d_in[13];
  const float* W_lin2     = (const float*)d_in[14];

  float* out = (float*)d_out;
  float* sc  = out + 2 * (size_t)N_NODES * MSG_DIM;

  float* h       = (float*)d_ws;
  float* msg_r   = h + (size_t)N_NODES * C_DIM;
  float* msg_i   = msg_r + (size_t)N_NODES * MSG_DIM;
  float* Wup_p   = msg_i + (size_t)N_NODES * MSG_DIM;
  float* Wskip_p = Wup_p + 128 * 128;          // 10 x [128,128]
  float* W1p     = Wskip_p + 10 * 128 * 128;
  float* W2p     = W1p + 8 * 64;
  float* W3p     = W2p + 64 * 64;
  float* W4p     = W3p + 64 * 64;
  float* L0p     = W4p + 64 * 384;
  float* L1p     = L0p + 128 * 128;
  float* L2p     = L1p + 128 * 128;

  // 0) zero atomic accumulators; fragment-pack all B-operand weights
  zero_kernel<<<2048, 256, 0, stream>>>(msg_r, 2 * (size_t)N_NODES * MSG_DIM);

  pack_b_kernel<<<64, 256, 0, stream>>>(W_up,   Wup_p, 128, 128, 128);
  for (int v = 0; v < A_DIM; ++v)
    pack_b_kernel<<<64, 256, 0, stream>>>(W_skip + v * C_DIM,
                                          Wskip_p + v * 128 * 128, 128, 128,
                                          A_DIM * C_DIM);
  pack_b_kernel<<<2,  256, 0, stream>>>(W_mlp1, W1p, 8,   64,  64);
  pack_b_kernel<<<16, 256, 0, stream>>>(W_mlp2, W2p, 64,  64,  64);
  pack_b_kernel<<<16, 256, 0, stream>>>(W_mlp3, W3p, 64,  64,  64);
  pack_b_kernel<<<96, 256, 0, stream>>>(W_mlp4, W4p, 64,  384, 384);
  pack_b_kernel<<<64, 256, 0, stream>>>(W_lin0, L0p, 128, 128, 128);
  pack_b_kernel<<<64, 256, 0, stream>>>(W_lin1, L1p, 128, 128, 128);
  pack_b_kernel<<<64, 256, 0, stream>>>(W_lin2, L2p, 128, 128, 128);

  // 1) node GEMMs: h and sc
  node_gemm_kernel<<<N_NODES / 16, 256, 0, stream>>>(
      node_attrs, node_feats, Wup_p, Wskip_p, h, sc);

  // 2) fused edge MLP + tensor product + atomic segment-sum
  edge_kernel<<<E_EDGES / 64, 128, 0, stream>>>(
      ear, eai, ef, eidx, W1p, W2p, W3p, W4p, h, msg_r, msg_i);

  // 3) irreps linear -> out_r, out_i
  linear_irreps_kernel<<<N_NODES / 16, 256, 0, stream>>>(
      msg_r, msg_i, L0p, L1p, L2p, out);
}